// CapsuleLayer_51694226375505
// MI455X (gfx1250) — compile-verified
//
#include <hip/hip_runtime.h>
#include <math.h>

#define IN_UNITS 1152
#define OUT_UNITS 32
#define CH 16
#define BATCH 64
#define N_ITER 3

typedef __attribute__((ext_vector_type(2))) float v2f;
typedef __attribute__((ext_vector_type(8))) float v8f;

// workspace layout: v[BATCH][OUT][CH][IN] floats, then R[BATCH][OUT][IN]
#define V_ELEMS ((size_t)BATCH * OUT_UNITS * CH * IN_UNITS)

// ---------------------------------------------------------------------------
// Kernel 1: v[b,i,o,x,z] = sum_y M[b,i,x,y] * W[i,o,y,z] via WMMA f32 16x16x4.
// Block = 1024 threads = 32 waves = 32 consecutive input units i for one
// (4-batch x 4-capsule) tile. Each wave does one WMMA:
//   A (16x4):  row m=(b,x), K=y.  VGPR0: K=0 (lanes 0-15) / K=2 (lanes 16-31)
//                                 VGPR1: K=1 / K=3
//   B (4x16):  col n=(o,z), K=y.  Same lane-half split of K.
//   D (16x16): vgpr r -> M = r (lanes 0-15) / r+8 (lanes 16-31), N = lane&15.
// D fragments are transposed through LDS so global stores are 128B-coalesced
// in the vws[b][o][c][i] layout (i fastest, matching the routing reads).
// ---------------------------------------------------------------------------
__global__ __launch_bounds__(1024)
void capsule_v_wmma(const float* __restrict__ M, const float* __restrict__ W,
                    float* __restrict__ vws) {
  const int wv   = threadIdx.x >> 5;   // wave in block == i offset (0..31)
  const int lane = threadIdx.x & 31;
  const int half = lane >> 4;
  const int l    = lane & 15;

  int bi = blockIdx.x;
  const int itile = bi % 36; bi /= 36;       // 1152/32 i-tiles
  const int otile = bi & 7;  bi >>= 3;       // 8 o-tiles
  const int btile = bi;                      // 16 b-tiles
  const int i  = itile * 32 + wv;
  const int b0 = btile * 4;
  const int o0 = otile * 4;

  __shared__ float lds[256 * 33];            // 256 rows x 32 (+1 pad)

  // A operand: row m = l -> (b, x)
  const int ab = b0 + (l >> 2);
  const int ax = l & 3;
  const float* mp = M + (((size_t)ab * IN_UNITS + i) * 4 + ax) * 4 + 2 * half;
  v2f A;
  A.x = mp[0];   // y = 2*half
  A.y = mp[1];   // y = 2*half+1

  // B operand: col n = l -> (o, z)
  const int bo = o0 + (l >> 2);
  const int bz = l & 3;
  const float* wp = W + (((size_t)i * OUT_UNITS + bo) * 4 + 2 * half) * 4 + bz;
  v2f B;
  B.x = wp[0];   // y = 2*half
  B.y = wp[4];   // y = 2*half+1

  v8f C = {};
  v8f D = __builtin_amdgcn_wmma_f32_16x16x4_f32(
      /*neg_a=*/false, A, /*neg_b=*/false, B,
      /*c_mod=*/(short)0, C, /*reuse_a=*/false, /*reuse_b=*/false);

  // Stage D to LDS: row = ((b-b0)*4 + (o-o0))*16 + c, c = x*4+z; col = wv (=ii)
#pragma unroll
  for (int r = 0; r < 8; ++r) {
    const int m   = r + 8 * half;
    const int dbp = m >> 2;        // b - b0
    const int dx  = m & 3;
    const int bop = l >> 2;        // o - o0
    const int row = (dbp * 4 + bop) * 16 + dx * 4 + bz;
    lds[row * 33 + wv] = D[r];
  }
  __syncthreads();

  // Coalesced flush: each 32-lane group writes one 128B row; 8 rounds = 256 rows
  const int ii = threadIdx.x & 31;
  const int r0 = threadIdx.x >> 5;
#pragma unroll
  for (int k = 0; k < 8; ++k) {
    const int row = r0 + 32 * k;               // 0..255
    const int db  = b0 + (row >> 6);
    const int oo  = o0 + ((row >> 4) & 3);
    const int cc  = row & 15;
    vws[(((size_t)db * OUT_UNITS + oo) * CH + cc) * IN_UNITS + itile * 32 + ii] =
        lds[row * 33 + ii];
  }
}

// ---------------------------------------------------------------------------
// Kernel 2: EM routing. One block per batch item; 32 waves = 32 out capsules.
// R kept in global workspace (L2-resident); per-capsule stats in LDS.
// ---------------------------------------------------------------------------
__global__ __launch_bounds__(1024)
void capsule_routing(const float* __restrict__ a_in,
                     const float* __restrict__ beta_u,
                     const float* __restrict__ beta_a,
                     const float* __restrict__ v,
                     float* __restrict__ Rg,
                     float* __restrict__ out) {
  const int b    = blockIdx.x;
  const int tid  = threadIdx.x;
  const int o    = tid >> 5;   // wave index == output capsule
  const int lane = tid & 31;

  __shared__ float mu_s[OUT_UNITS][CH];
  __shared__ float i2v_s[OUT_UNITS][CH];   // 1/(2*var)
  __shared__ float aeff_s[OUT_UNITS];      // a_o * prod(2*pi*var)

  float*       Rb = Rg + (size_t)b * OUT_UNITS * IN_UNITS;
  const float* vb = v  + (size_t)b * OUT_UNITS * CH * IN_UNITS;
  const float* av = a_in + (size_t)b * IN_UNITS;

  // R = 1/OUT_UNITS
  for (int idx = tid; idx < OUT_UNITS * IN_UNITS; idx += 1024)
    Rb[idx] = 1.0f / OUT_UNITS;
  __syncthreads();

  const float lambs[N_ITER] = {0.01f, 0.012f, 0.0144f};  // 0.01 * 1.2^it

  for (int it = 0; it < N_ITER; ++it) {
    // ---- pass 1: weighted moments over i for this wave's capsule o ----
    float sumR = 0.0f;
    float s1[CH], s2[CH];
#pragma unroll
    for (int c = 0; c < CH; ++c) { s1[c] = 0.0f; s2[c] = 0.0f; }

    const float* vo = vb + (size_t)o * CH * IN_UNITS;
    const float* Ro = Rb + o * IN_UNITS;
    for (int i = lane; i < IN_UNITS; i += 32) {
      const float rw = Ro[i] * av[i];       // R *= input_a (on the fly)
      sumR += rw;
#pragma unroll
      for (int c = 0; c < CH; ++c) {
        const float val = vo[c * IN_UNITS + i];
        s1[c] += rw * val;
        s2[c] += rw * val * val;
      }
    }
    // butterfly reduction: every lane ends with the full sums
#pragma unroll
    for (int msk = 16; msk >= 1; msk >>= 1) {
      sumR += __shfl_xor(sumR, msk, 32);
#pragma unroll
      for (int c = 0; c < CH; ++c) {
        s1[c] += __shfl_xor(s1[c], msk, 32);
        s2[c] += __shfl_xor(s2[c], msk, 32);
      }
    }

    if (lane == 0) {
      const float inv = 1.0f / sumR;
      float logsum = 0.0f;
      float eff    = 1.0f;
      float mu[CH];
#pragma unroll
      for (int c = 0; c < CH; ++c) {
        mu[c] = s1[c] * inv;
        float var = s2[c] * inv - mu[c] * mu[c];  // E[v^2]-E[v]^2
        logsum += 0.5f * __logf(var);             // log(sigma)
        eff    *= 2.0f * 3.14159265358979323846f * var;
        mu_s[o][c]  = mu[c];
        i2v_s[o][c] = 0.5f / var;
      }
      const float costSum = sumR * ((float)CH * beta_u[o] + logsum);
      const float z   = lambs[it] * (beta_a[o] - costSum);
      const float a_o = 1.0f / (1.0f + __expf(-z));
      aeff_s[o] = a_o * eff;
      if (it == N_ITER - 1) {
        out[b * OUT_UNITS + o] = a_o;
#pragma unroll
        for (int c = 0; c < CH; ++c)
          out[BATCH * OUT_UNITS + (b * OUT_UNITS + o) * CH + c] = mu[c];
      }
    }
    __syncthreads();
    if (it == N_ITER - 1) break;

    // ---- pass 2: numer[o][i] = a_o * eff * exp(-sum_c (v-mu)^2/(2 var)) ----
    for (int idx = tid; idx < OUT_UNITS * IN_UNITS; idx += 1024) {
      const int oo = idx / IN_UNITS;
      const int i  = idx - oo * IN_UNITS;
      const float* vp = vb + (size_t)oo * CH * IN_UNITS + i;
      float ex = 0.0f;
#pragma unroll
      for (int c = 0; c < CH; ++c) {
        const float d = vp[c * IN_UNITS] - mu_s[oo][c];
        ex += d * d * i2v_s[oo][c];
      }
      Rb[idx] = aeff_s[oo] * __expf(-ex);
    }
    __syncthreads();

    // ---- pass 3: normalize over output capsules ----
    for (int i = tid; i < IN_UNITS; i += 1024) {
      float s = 0.0f;
#pragma unroll
      for (int oo = 0; oo < OUT_UNITS; ++oo) s += Rb[oo * IN_UNITS + i];
      const float inv = 1.0f / s;
#pragma unroll
      for (int oo = 0; oo < OUT_UNITS; ++oo) Rb[oo * IN_UNITS + i] *= inv;
    }
    __syncthreads();
  }
}

extern "C" void kernel_launch(void* const* d_in, const int* in_sizes, int n_in,
                              void* d_out, int out_size, void* d_ws, size_t ws_size,
                              hipStream_t stream) {
  const float* input_a = (const float*)d_in[0];   // (64,1152)
  const float* input_M = (const float*)d_in[1];   // (64,1152,4,4)
  const float* W       = (const float*)d_in[2];   // (1152,32,4,4)
  const float* beta_u  = (const float*)d_in[3];   // (32,)
  const float* beta_a  = (const float*)d_in[4];   // (32,)
  float* out = (float*)d_out;

  float* vws = (float*)d_ws;            // 151 MB
  float* Rg  = vws + V_ELEMS;           // +9.4 MB

  // 16 b-tiles * 8 o-tiles * 36 i-tiles; 32 waves (one WMMA each) per block
  const int gridV = 16 * 8 * (IN_UNITS / 32);
  capsule_v_wmma<<<gridV, 1024, 0, stream>>>(input_M, W, vws);
  capsule_routing<<<BATCH, 1024, 0, stream>>>(input_a, beta_u, beta_a, vws, Rg, out);
}